// SimLoss_10892037062972
// MI455X (gfx1250) — compile-verified
//
#include <hip/hip_runtime.h>

typedef __attribute__((ext_vector_type(2))) float v2f;
typedef __attribute__((ext_vector_type(8))) float v8f;

#define SIM_N 8192
#define SIM_M 128
static constexpr float kEps = 1e-4f;

__device__ __forceinline__ void onlineMerge(float& m, float& s, float m2, float s2) {
    float mm = fmaxf(m, m2);
    s = s * __expf(m - mm) + s2 * __expf(m2 - mm);
    m = mm;
}

// Per-row online-softmax stats (max, sum of exp(s-max)) of S = X @ Y^T,
// S[i][j] = dot(X[i,:], Y[j,:]).  One wave per 16-row strip, sweeping all
// 512 column tiles with V_WMMA_F32_16X16X4_F32 (K-loop of 32).
__global__ void __launch_bounds__(256)
simloss_stats_kernel(const float* __restrict__ X, const float* __restrict__ Y,
                     float* __restrict__ rmax, float* __restrict__ rsum) {
    const int lane = threadIdx.x & 31;
    const int wave = blockIdx.x * (blockDim.x >> 5) + (threadIdx.x >> 5);
    const int rowb = wave * 16;
    const int mrow = lane & 15;          // M (for A) / N (for B) index
    const int ksub = (lane >> 4) << 1;   // K sub-offset {0,2} per lane half
    const int half = lane >> 4;

    // Preload A strip: 16 rows x 128 K  ->  32 x v2f per lane (reused 512x)
    v2f a[32];
    const float* xp = X + (rowb + mrow) * SIM_M + ksub;
    #pragma unroll
    for (int t = 0; t < 32; ++t) { a[t].x = xp[4 * t]; a[t].y = xp[4 * t + 1]; }

    float runm[8], runs[8];
    #pragma unroll
    for (int r = 0; r < 8; ++r) { runm[r] = -3.4e38f; runs[r] = 0.f; }

    for (int jb = 0; jb < SIM_N; jb += 16) {
        const float* yp = Y + (jb + mrow) * SIM_M + ksub;
        v8f c = {};
        #pragma unroll
        for (int t = 0; t < 32; ++t) {
            v2f b; b.x = yp[4 * t]; b.y = yp[4 * t + 1];
            // D = A x B + C, fp32 16x16x4
            c = __builtin_amdgcn_wmma_f32_16x16x4_f32(
                    false, a[t], false, b, (short)0, c, false, false);
        }
        // c[r] = S[rowb + 8*half + r][jb + mrow] : per-lane online update
        #pragma unroll
        for (int r = 0; r < 8; ++r) {
            float v = c[r];
            float mm = fmaxf(runm[r], v);
            runs[r] = runs[r] * __expf(runm[r] - mm) + __expf(v - mm);
            runm[r] = mm;
        }
    }

    // Merge the 16 column-lanes inside each half (rows stay per-half)
    #pragma unroll
    for (int off = 1; off < 16; off <<= 1) {
        #pragma unroll
        for (int r = 0; r < 8; ++r) {
            float om = __shfl_xor(runm[r], off, 32);
            float os = __shfl_xor(runs[r], off, 32);
            onlineMerge(runm[r], runs[r], om, os);
        }
    }
    if (mrow == 0) {
        #pragma unroll
        for (int r = 0; r < 8; ++r) {
            rmax[rowb + half * 8 + r] = runm[r];
            rsum[rowb + half * 8 + r] = runs[r];
        }
    }
}

// Recompute each 16x16 tile of S, stream label (the only HBM-bound input),
// accumulate -label * (log(p_row+eps) + log(p_col+eps)).  One partial/wave.
__global__ void __launch_bounds__(256)
simloss_loss_kernel(const float* __restrict__ X /*out2*/,
                    const float* __restrict__ Y /*out1*/,
                    const float* __restrict__ label,
                    const float* __restrict__ rmax, const float* __restrict__ rsum,
                    const float* __restrict__ cmax, const float* __restrict__ csum,
                    float* __restrict__ partials) {
    const int lane = threadIdx.x & 31;
    const int wave = blockIdx.x * (blockDim.x >> 5) + (threadIdx.x >> 5);
    const int rowb = wave * 16;
    const int mrow = lane & 15;
    const int ksub = (lane >> 4) << 1;
    const int half = lane >> 4;

    v2f a[32];
    const float* xp = X + (rowb + mrow) * SIM_M + ksub;
    #pragma unroll
    for (int t = 0; t < 32; ++t) { a[t].x = xp[4 * t]; a[t].y = xp[4 * t + 1]; }

    float rm[8], rinv[8];
    #pragma unroll
    for (int r = 0; r < 8; ++r) {
        int row = rowb + half * 8 + r;
        rm[r]   = rmax[row];
        rinv[r] = 1.0f / rsum[row];
    }

    float acc = 0.f;
    for (int jb = 0; jb < SIM_N; jb += 16) {
        const float* yp = Y + (jb + mrow) * SIM_M + ksub;
        v8f c = {};
        #pragma unroll
        for (int t = 0; t < 32; ++t) {
            v2f b; b.x = yp[4 * t]; b.y = yp[4 * t + 1];
            c = __builtin_amdgcn_wmma_f32_16x16x4_f32(
                    false, a[t], false, b, (short)0, c, false, false);
        }
        const int   col  = jb + mrow;
        const float cm   = cmax[col];
        const float cinv = 1.0f / csum[col];
        #pragma unroll
        for (int r = 0; r < 8; ++r) {
            const int   row = rowb + half * 8 + r;
            const float s   = c[r];
            const float pr  = __expf(s - rm[r]) * rinv[r];
            const float pc  = __expf(s - cm) * cinv;
            acc -= label[(size_t)row * SIM_N + col] *
                   (__logf(pr + kEps) + __logf(pc + kEps));
        }
    }
    #pragma unroll
    for (int off = 16; off >= 1; off >>= 1)
        acc += __shfl_xor(acc, off, 32);
    if (lane == 0) partials[wave] = acc;
}

// Deterministic final reduction of the 512 wave partials (no float atomics,
// so graph replays are bit-identical).
__global__ void __launch_bounds__(256)
simloss_reduce_kernel(const float* __restrict__ partials, float* __restrict__ out) {
    __shared__ float sm[256];
    const int t = threadIdx.x;
    sm[t] = partials[t] + partials[t + 256];
    __syncthreads();
    for (int s = 128; s > 0; s >>= 1) {
        if (t < s) sm[t] += sm[t + s];
        __syncthreads();
    }
    if (t == 0) out[0] = sm[0];
}

extern "C" void kernel_launch(void* const* d_in, const int* in_sizes, int n_in,
                              void* d_out, int out_size, void* d_ws, size_t ws_size,
                              hipStream_t stream) {
    const float* out1  = (const float*)d_in[0];
    const float* out2  = (const float*)d_in[1];
    const float* label = (const float*)d_in[2];

    float* ws       = (float*)d_ws;
    float* rmax     = ws;                 // [8192]
    float* rsum     = ws + SIM_N;         // [8192]
    float* cmax     = ws + 2 * SIM_N;     // [8192]
    float* csum     = ws + 3 * SIM_N;     // [8192]
    float* partials = ws + 4 * SIM_N;     // [512]

    const dim3 blk(256);                       // 8 wave32 per block
    const dim3 grd(SIM_N / 16 / 8);            // 64 blocks -> 512 waves

    // Row stats of S = out2 @ out1^T
    simloss_stats_kernel<<<grd, blk, 0, stream>>>(out2, out1, rmax, rsum);
    // Row stats of S^T  == column stats of S
    simloss_stats_kernel<<<grd, blk, 0, stream>>>(out1, out2, cmax, csum);
    // Fused recompute + label stream + loss accumulate
    simloss_loss_kernel<<<grd, blk, 0, stream>>>(out2, out1, label,
                                                 rmax, rsum, cmax, csum, partials);
    simloss_reduce_kernel<<<1, 256, 0, stream>>>(partials, (float*)d_out);
}